// SinkhornDistance_75256416960636
// MI455X (gfx1250) — compile-verified
//
#include <hip/hip_runtime.h>
#include <hip/hip_bf16.h>

typedef float v2f __attribute__((ext_vector_type(2)));
typedef float v8f __attribute__((ext_vector_type(8)));

#define BATCH 4
#define P 3072
#define ROWS (BATCH * P)                 // 12288
#define NELEM ((size_t)BATCH * P * P)    // 37748736

// workspace layout (float indices)
#define WS_U    0
#define WS_V    (WS_U + ROWS)
#define WS_X2   (WS_V + ROWS)
#define WS_Y2   (WS_X2 + ROWS)
#define WS_ERR  (WS_Y2 + ROWS)           // u64 (occupies 2 float slots, 8B aligned)
#define WS_DONE (WS_ERR + 2)             // int

#define INV_EPS 10.0f
#define EPSF    0.1f
#define LOG_MU  (-8.0300534f)            // log(1/3072 + 1e-8)
#define ERR_SCALE 16777216.0f            // 2^24
#define ERR_DENOM 67108864.0             // 4 * 2^24  (mean over batch)

__device__ __forceinline__ void lse_update(float& m, float& s, float val) {
    if (val <= m) {
        s += __expf(val - m);
    } else {
        s = s * __expf(m - val) + 1.0f;
        m = val;
    }
}

// ---------------- init: x2/y2 row norms, zero u/v/err/done/cost ----------------
__global__ void kInit(const float* __restrict__ x, const float* __restrict__ y,
                      float* __restrict__ ws, float* __restrict__ cost) {
    int idx = blockIdx.x * blockDim.x + threadIdx.x;  // 0..24575
    if (idx < ROWS) {
        const float* p = x + (size_t)idx * 32;
        float s = 0.f;
#pragma unroll
        for (int k = 0; k < 32; ++k) s += p[k] * p[k];
        ws[WS_X2 + idx] = s;
        ws[WS_U + idx] = 0.f;
    } else {
        int j = idx - ROWS;
        const float* p = y + (size_t)j * 32;
        float s = 0.f;
#pragma unroll
        for (int k = 0; k < 32; ++k) s += p[k] * p[k];
        ws[WS_Y2 + j] = s;
        ws[WS_V + j] = 0.f;
    }
    if (idx == 0) {
        *(unsigned long long*)(ws + WS_ERR) = 0ull;
        *(int*)(ws + WS_DONE) = 0;
        cost[0] = 0.f; cost[1] = 0.f; cost[2] = 0.f; cost[3] = 0.f;
    }
}

// ---------------- C = x2 + y2 - 2 x.y^T via f32 WMMA (one wave per 16x16 tile) ----------------
__global__ void kC(const float* __restrict__ x, const float* __restrict__ y,
                   const float* __restrict__ ws, float* __restrict__ Cout) {
    const int TILES_J = P / 16;             // 192
    const int TILES_PER_B = TILES_J * TILES_J;
    int wid  = blockIdx.x * (blockDim.x >> 5) + (threadIdx.x >> 5);
    int lane = threadIdx.x & 31;

    int b  = wid / TILES_PER_B;
    int t  = wid % TILES_PER_B;
    int i0 = (t / TILES_J) * 16;
    int j0 = (t % TILES_J) * 16;

    const float* xb = x + (size_t)b * P * 32;
    const float* yb = y + (size_t)b * P * 32;

    // ISA f32 A/B 16x4 layout: lanes 0-15 hold K=0,1 (v0,v1); lanes 16-31 hold K=2,3.
    int m  = lane & 15;
    int kk = (lane >> 4) << 1;              // 0 or 2
    const float* xrow = xb + (size_t)(i0 + m) * 32 + kk;
    const float* yrow = yb + (size_t)(j0 + m) * 32 + kk;

    v8f acc = {0.f, 0.f, 0.f, 0.f, 0.f, 0.f, 0.f, 0.f};
#pragma unroll
    for (int k0 = 0; k0 < 32; k0 += 4) {
        v2f a  = *(const v2f*)(xrow + k0);
        v2f bb = *(const v2f*)(yrow + k0);
        acc = __builtin_amdgcn_wmma_f32_16x16x4_f32(
            false, a, false, bb, (short)0, acc, false, false);
    }

    // C/D layout: VGPR r, lanes 0-15 -> M=r; lanes 16-31 -> M=r+8; N = lane&15
    int col   = j0 + (lane & 15);
    int rbase = i0 + ((lane >> 4) << 3);
    float y2c = ws[WS_Y2 + b * P + col];
    size_t cb = (size_t)b * P * P;
#pragma unroll
    for (int r = 0; r < 8; ++r) {
        int row = rbase + r;
        float val = ws[WS_X2 + b * P + row] + y2c - 2.0f * acc[r];
        Cout[cb + (size_t)row * P + col] = val;
    }
}

// ---------------- row pass: u update (one wave per row) ----------------
__global__ void kRow(const float* __restrict__ C, float* __restrict__ ws) {
    if (*(const int*)(ws + WS_DONE)) return;
    float* u = ws + WS_U;
    const float* v = ws + WS_V;

    int w    = blockIdx.x * (blockDim.x >> 5) + (threadIdx.x >> 5);  // global row 0..12287
    int lane = threadIdx.x & 31;
    int b    = w / P;
    const float* crow = C + (size_t)w * P;
    const float* vb   = v + b * P;
    float ui = u[w];

    float m = -3.4e38f, s = 0.f;
    for (int j = lane; j < P; j += 32) {
        float val = (ui + vb[j] - crow[j]) * INV_EPS;
        lse_update(m, s, val);
    }
#pragma unroll
    for (int off = 16; off > 0; off >>= 1) {
        float om = __shfl_xor(m, off, 32);
        float os = __shfl_xor(s, off, 32);
        float nm = fmaxf(m, om);
        s = s * __expf(m - nm) + os * __expf(om - nm);
        m = nm;
    }
    if (lane == 0) {
        float lse = m + logf(s);
        float un  = EPSF * (LOG_MU - lse) + ui;
        u[w] = un;
        unsigned long long q = (unsigned long long)(fabsf(un - ui) * ERR_SCALE);
        atomicAdd((unsigned long long*)(ws + WS_ERR), q);
    }
}

// ---------------- col pass: v update (32 cols x 8 i-streams per block) ----------------
__global__ void kCol(const float* __restrict__ C, float* __restrict__ ws) {
    if (*(const int*)(ws + WS_DONE)) return;
    const float* u = ws + WS_U;
    float* v = ws + WS_V;

    int b    = blockIdx.x / (P / 32);            // 96 col-tiles per batch
    int j0   = (blockIdx.x % (P / 32)) * 32;
    int col  = threadIdx.x & 31;
    int strm = threadIdx.x >> 5;                 // 0..7
    int j    = j0 + col;

    const float* ub = u + b * P;
    float vj = v[b * P + j];
    const float* cb = C + (size_t)b * P * P;

    float m = -3.4e38f, s = 0.f;
    for (int i = strm; i < P; i += 8) {
        float val = (ub[i] + vj - cb[(size_t)i * P + j]) * INV_EPS;
        lse_update(m, s, val);
    }

    __shared__ float sm[8][32];
    __shared__ float ss[8][32];
    sm[strm][col] = m;
    ss[strm][col] = s;
    __syncthreads();
    if (threadIdx.x < 32) {
        float M = sm[0][col], S = ss[0][col];
#pragma unroll
        for (int t = 1; t < 8; ++t) {
            float om = sm[t][col], os = ss[t][col];
            float nm = fmaxf(M, om);
            S = S * __expf(M - nm) + os * __expf(om - nm);
            M = nm;
        }
        float lse = M + logf(S);
        v[b * P + j] = EPSF * (LOG_MU - lse) + vj;
    }
}

// ---------------- done-flag update (deterministic fixed-point err) ----------------
__global__ void kDone(float* __restrict__ ws) {
    unsigned long long* err = (unsigned long long*)(ws + WS_ERR);
    int* done = (int*)(ws + WS_DONE);
    if (!*done) {
        double e = (double)(*err) / ERR_DENOM;
        if (e < 0.1) *done = 1;
    }
    *err = 0ull;
}

// ---------------- final: pi = exp(M), cost[b] = sum(pi * C) ----------------
__global__ void kPi(const float* __restrict__ C, const float* __restrict__ ws,
                    float* __restrict__ pi, float* __restrict__ cost) {
    const float* u = ws + WS_U;
    const float* v = ws + WS_V;
    const int BLOCKS_PER_B = (P * P) / 256;      // 36864
    int b  = blockIdx.x / BLOCKS_PER_B;
    int bb = blockIdx.x % BLOCKS_PER_B;
    int i  = bb / (P / 256);                     // 12 blocks per row
    int j  = (bb % (P / 256)) * 256 + threadIdx.x;

    size_t idx = ((size_t)b * P + i) * P + j;
    float Cv = C[idx];
    float p  = expf((u[b * P + i] + v[b * P + j] - Cv) * INV_EPS);
    pi[idx] = p;

    float contrib = p * Cv;
#pragma unroll
    for (int off = 16; off > 0; off >>= 1)
        contrib += __shfl_xor(contrib, off, 32);
    __shared__ float red[8];
    if ((threadIdx.x & 31) == 0) red[threadIdx.x >> 5] = contrib;
    __syncthreads();
    if (threadIdx.x == 0) {
        float tot = 0.f;
#pragma unroll
        for (int t = 0; t < 8; ++t) tot += red[t];
        atomicAdd(&cost[b], tot);
    }
}

extern "C" void kernel_launch(void* const* d_in, const int* in_sizes, int n_in,
                              void* d_out, int out_size, void* d_ws, size_t ws_size,
                              hipStream_t stream) {
    const float* x = (const float*)d_in[0];
    const float* y = (const float*)d_in[1];
    float* out  = (float*)d_out;
    float* cost = out;                       // [4]
    float* pi   = out + BATCH;               // [4,3072,3072]
    float* Cbuf = out + BATCH + NELEM;       // [4,3072,3072]
    float* ws   = (float*)d_ws;

    kInit<<<96, 256, 0, stream>>>(x, y, ws, cost);
    kC<<<18432, 256, 0, stream>>>(x, y, ws, Cbuf);   // 4*192*192 tiles, 8 waves/block
    for (int it = 0; it < 50; ++it) {
        kRow<<<1536, 256, 0, stream>>>(Cbuf, ws);    // 12288 rows, wave per row
        kCol<<<384, 256, 0, stream>>>(Cbuf, ws);     // 4*96 col-tiles
        kDone<<<1, 1, 0, stream>>>(ws);
    }
    kPi<<<147456, 256, 0, stream>>>(Cbuf, ws, pi, cost);
}